// AttentionLayer_29317446763280
// MI455X (gfx1250) — compile-verified
//
#include <hip/hip_runtime.h>
#include <hip/hip_bf16.h>

// ---------------------------------------------------------------------------
// Bahdanau additive attention, fused for MI455X (gfx1250, wave32).
//
// Stage 1/2: fp32 projections via V_WMMA_F32_16X16X4_F32.
//   one  [B*Td=1024, 256] = dec @ W_dec            (row-major)
//   twoT [B, 256(d), 256(s)] = (enc @ W_enc)^T     (transposed per-batch so
//                                                   the score loop is coalesced)
// Stage 3: fused score + softmax, no [B,Td,Te,D] tensor ever materialized.
// ---------------------------------------------------------------------------

typedef float v2f __attribute__((ext_vector_type(2)));
typedef float v8f __attribute__((ext_vector_type(8)));

#define D_DIM 256

// One wave (32 lanes) computes one 16x16 output tile; blockDim = (32,4) so a
// block covers a 16x64 strip of the output (4 waves, 4 N-tiles).
// X: [M x 256] row-major, W: [256 x 256] row-major, Y = X @ W.
// transpose_out: store Y[row, col] at twoT[(b*256 + col)*256 + s] with
//                row = b*256 + s  (Te == D == 256 makes this exact).
__global__ void __launch_bounds__(128)
proj_gemm_wmma(const float* __restrict__ X, const float* __restrict__ W,
               float* __restrict__ Y, int transpose_out) {
    const int lane = threadIdx.x;          // 0..31
    const int half = lane >> 4;            // 0: K pair {0,1}, 1: K pair {2,3}
    const int l    = lane & 15;            // M (for A) / N (for B,C,D)
    const int row0 = blockIdx.x * 16;
    const int col0 = (blockIdx.y * 4 + threadIdx.y) * 16;

    v8f acc = {};
    #pragma unroll 4
    for (int kb = 0; kb < D_DIM; kb += 4) {
        const int k0 = kb + 2 * half;
        // A fragment: 16x4, lane = M, v0/v1 = consecutive K within half
        v2f a;
        a.x = X[(row0 + l) * D_DIM + k0];
        a.y = X[(row0 + l) * D_DIM + k0 + 1];
        // B fragment: 4x16, lane = N, same K striping (coalesced over lanes)
        v2f b;
        b.x = W[(k0)     * D_DIM + col0 + l];
        b.y = W[(k0 + 1) * D_DIM + col0 + l];
        // (neg_a, A, neg_b, B, c_mod, C, reuse_a, reuse_b)
        acc = __builtin_amdgcn_wmma_f32_16x16x4_f32(
            false, a, false, b, (short)0, acc, false, false);
    }

    // C/D layout: VGPR r -> M = r + 8*half, N = l
    if (!transpose_out) {
        #pragma unroll
        for (int r = 0; r < 8; ++r) {
            Y[(row0 + r + 8 * half) * D_DIM + col0 + l] = acc[r];
        }
    } else {
        #pragma unroll
        for (int r = 0; r < 8; ++r) {
            const int grow = row0 + r + 8 * half;   // b*256 + s
            const int b    = grow >> 8;
            const int s    = grow & 255;
            const int d    = col0 + l;
            Y[((b << 8) + d) * 256 + s] = acc[r];
        }
    }
}

__device__ __forceinline__ float fast_tanh(float x) {
    // tanh(x) = (e^{2x} - 1) / (e^{2x} + 1); clamp so exp never overflows
    // (tanh(+-9) == +-1 at fp32 precision).
    x = fminf(fmaxf(x, -9.0f), 9.0f);
    const float e = __expf(2.0f * x);                 // v_exp_f32
    return (e - 1.0f) * __builtin_amdgcn_rcpf(e + 1.0f);
}

// One block per (b, t): 256 threads, thread = encoder step s.
// score[b,t,s] = sum_d w[d] * tanh(one[b,t,d] + twoT[b,d,s]), then softmax
// over s. twoT layout makes the inner load coalesced across lanes for every d.
__global__ void __launch_bounds__(256)
score_softmax(const float* __restrict__ one, const float* __restrict__ twoT,
              const float* __restrict__ w,
              float* __restrict__ out_score, float* __restrict__ out_wts) {
    __shared__ float s_one[D_DIM];
    __shared__ float s_w[D_DIM];
    __shared__ float red[256];

    const int bt  = blockIdx.x;       // b*128 + t
    const int b   = bt >> 7;
    const int tid = threadIdx.x;      // s

    s_one[tid] = one[bt * D_DIM + tid];
    s_w[tid]   = w[tid];
    __syncthreads();

    const float* tptr = twoT + (size_t)(b << 8) * 256;  // [d][s] for batch b
    float acc = 0.0f;
    #pragma unroll 8
    for (int d = 0; d < D_DIM; ++d) {
        acc += s_w[d] * fast_tanh(s_one[d] + tptr[d * 256 + tid]);
    }
    out_score[bt * 256 + tid] = acc;

    // ---- numerically stable softmax over the 256 encoder steps ----
    red[tid] = acc;
    __syncthreads();
    #pragma unroll
    for (int off = 128; off > 0; off >>= 1) {
        if (tid < off) red[tid] = fmaxf(red[tid], red[tid + off]);
        __syncthreads();
    }
    const float m = red[0];
    __syncthreads();

    const float e = __expf(acc - m);
    red[tid] = e;
    __syncthreads();
    #pragma unroll
    for (int off = 128; off > 0; off >>= 1) {
        if (tid < off) red[tid] += red[tid + off];
        __syncthreads();
    }
    const float inv_sum = 1.0f / red[0];

    out_wts[bt * 256 + tid] = e * inv_sum;
}

extern "C" void kernel_launch(void* const* d_in, const int* in_sizes, int n_in,
                              void* d_out, int out_size, void* d_ws, size_t ws_size,
                              hipStream_t stream) {
    (void)in_sizes; (void)n_in; (void)out_size; (void)ws_size;
    const float* enc = (const float*)d_in[0];   // [8, 256, 256]
    const float* dec = (const float*)d_in[1];   // [8, 128, 256]
    const float* Wd  = (const float*)d_in[2];   // [256, 256]
    const float* We  = (const float*)d_in[3];   // [256, 256]
    const float* Wc  = (const float*)d_in[4];   // [256, 1]

    float* out   = (float*)d_out;               // score (262144) | wts (262144)
    float* one   = (float*)d_ws;                // 1024 * 256 floats (1 MB)
    float* twoT  = one + 1024 * 256;            // 2048 * 256 floats (2 MB)

    // one = dec @ W_dec  : M = 1024 -> grid (64, 4), 128 threads (4 waves)
    proj_gemm_wmma<<<dim3(64, 4), dim3(32, 4), 0, stream>>>(dec, Wd, one, 0);
    // twoT = (enc @ W_enc)^T per batch : M = 2048 -> grid (128, 4)
    proj_gemm_wmma<<<dim3(128, 4), dim3(32, 4), 0, stream>>>(enc, We, twoT, 1);
    // fused score + softmax: 1024 blocks of 256 threads
    score_softmax<<<1024, 256, 0, stream>>>(one, twoT, Wc, out, out + 262144);
}